// PointTransformerLayer_16630113370720
// MI455X (gfx1250) — compile-verified
//
#include <hip/hip_runtime.h>
#include <hip/hip_bf16.h>
#include <math.h>

// ---------------------------------------------------------------------------
// PointTransformerLayer fused implementation for gfx1250 (CDNA5, wave32).
// fp32 end-to-end using V_WMMA_F32_16X16X4_F32 to preserve reference numerics.
//
// Weight matrices live in LDS in a pair-interleaved, XOR-swizzled layout so
// the WMMA B operand {W[k][n], W[k+1][n]} is a single conflict-free
// ds_load_b64 with an immediate offset (no per-step address math, no operand
// marshalling movs).
// ---------------------------------------------------------------------------

typedef float v2f __attribute__((ext_vector_type(2)));
typedef float v8f __attribute__((ext_vector_type(8)));

#define CCH   64          // channels
#define KNBR  16          // neighbors per point
#define STG   68          // staging row stride in floats (bank-conflict-free, 8B aligned)
#define WAVES_PER_BLOCK 8

__device__ __forceinline__ v8f wmma_f32(v2f a, v2f b, v8f c) {
  // 8-arg VOP3P form: (neg_a, A, neg_b, B, c_mod, C, reuse_a, reuse_b)
  return __builtin_amdgcn_wmma_f32_16x16x4_f32(false, a, false, b, (short)0, c,
                                               false, false);
}

// Swizzled LDS index for weight element pair-row pr = k>>1, column n.
//  - pairs (k, k+1) interleaved -> one ds_load_b64 per B operand
//  - XOR of the tile bit for odd pair-rows -> the two half-waves use disjoint
//    32-bank sets (conflict-free b64 loads)
__device__ __forceinline__ int bswz(int pr, int n) {
  return pr * 128 + ((n ^ ((pr & 1) << 4)) << 1);
}

// Cooperative copy of a 64x64 weight matrix into swizzled LDS layout.
__device__ __forceinline__ void load_w_swz(float* dst, const float* __restrict__ src) {
  for (int idx = threadIdx.x; idx < CCH * CCH; idx += 256) {
    int k = idx >> 6, n = idx & 63;
    dst[bswz(k >> 1, n) + (k & 1)] = src[idx];
  }
}

// ---------------------------------------------------------------------------
// Kernel 1: q/k/v = features @ W{q,k,v} + b.  One wave per 16-row tile.
// A operand preloaded in WMMA A-layout straight from global (8B loads).
// ---------------------------------------------------------------------------
__global__ __launch_bounds__(256) void qkv_kernel(
    const float* __restrict__ features,
    const float* __restrict__ Wq, const float* __restrict__ bq,
    const float* __restrict__ Wk, const float* __restrict__ bk,
    const float* __restrict__ Wv, const float* __restrict__ bv,
    float* __restrict__ qb, float* __restrict__ kb, float* __restrict__ vb,
    int ntiles)
{
  __shared__ float ldsW[3 * CCH * CCH];
  load_w_swz(ldsW,               Wq);
  load_w_swz(ldsW + CCH * CCH,   Wk);
  load_w_swz(ldsW + 2 * CCH * CCH, Wv);
  __syncthreads();

  int wave = blockIdx.x * WAVES_PER_BLOCK + (threadIdx.x >> 5);
  if (wave >= ntiles) return;                      // wave-uniform branch
  int lane = threadIdx.x & 31;
  int hi = lane >> 4, lm = lane & 15;
  int base = wave * 16;

  // A-layout (f32 16x4): lanes 0-15 hold {k0,k0+1} of row lm, lanes 16-31 {k0+2,k0+3}
  v2f A[16];
  const float* arow = features + (size_t)(base + lm) * CCH;
#pragma unroll
  for (int s = 0; s < 16; ++s)
    A[s] = *(const v2f*)(arow + 4 * s + 2 * hi);

  const float* Bias[3] = {bq, bk, bv};
  float*       Out[3]  = {qb, kb, vb};

#pragma unroll
  for (int w = 0; w < 3; ++w) {
    const float* Wl = ldsW + w * CCH * CCH;
#pragma unroll
    for (int t = 0; t < 4; ++t) {
      // per-tile B base: pr = 2s+hi -> addr = base + s*256 dwords (imm offsets)
      const float* bbase = Wl + hi * 128 + (((16 * t + lm) ^ (hi << 4)) << 1);
      v8f acc = {};
#pragma unroll
      for (int s = 0; s < 16; ++s) {
        v2f b = *(const v2f*)(bbase + s * 256);    // ds_load_b64, imm offset
        acc = wmma_f32(A[s], b, acc);
      }
      float bias = Bias[w][16 * t + lm];
      float* op  = Out[w];
#pragma unroll
      for (int r = 0; r < 8; ++r)
        op[(size_t)(base + r + 8 * hi) * CCH + 16 * t + lm] = acc[r] + bias;
    }
  }
}

// ---------------------------------------------------------------------------
// Kernel 2: fused neighborhood transformer, one wave32 per point.
//   p    = relu(bn(rel @ Wp + b_p))            (4 WMMAs, K=3 zero-padded to 4)
//   a    = q_i - gather(k) + p  -> LDS staging
//   t1   = relu(bn(a @ Wg1 + bg1))             (64 WMMAs)
//   h    = t1 @ Wg2 + bg2                      (64 WMMAs)
//   attn = softmax over the 16 neighbors (in-lane + shfl_xor 16)
//   s    = sum_m (gather(v)+p)*attn            -> workspace
// ---------------------------------------------------------------------------
__global__ __launch_bounds__(256) void pt_main_kernel(
    const float* __restrict__ points, const int* __restrict__ gidx,
    const float* __restrict__ qb, const float* __restrict__ kb,
    const float* __restrict__ vb,
    const float* __restrict__ Wp,   const float* __restrict__ b_p,
    const float* __restrict__ bnp_g, const float* __restrict__ bnp_b,
    const float* __restrict__ bnp_m, const float* __restrict__ bnp_v,
    const float* __restrict__ Wg1,  const float* __restrict__ bg1,
    const float* __restrict__ bng_g, const float* __restrict__ bng_b,
    const float* __restrict__ bng_m, const float* __restrict__ bng_v,
    const float* __restrict__ Wg2,  const float* __restrict__ bg2,
    float* __restrict__ sb, int N)
{
  __shared__ float w1[CCH * CCH];
  __shared__ float w2[CCH * CCH];
  __shared__ float stg[WAVES_PER_BLOCK * 16 * STG];
  load_w_swz(w1, Wg1);
  load_w_swz(w2, Wg2);
  __syncthreads();

  int winb = threadIdx.x >> 5;
  int i    = blockIdx.x * WAVES_PER_BLOCK + winb;
  if (i >= N) return;                              // wave-uniform branch
  int lane = threadIdx.x & 31;
  int hi = lane >> 4, lm = lane & 15;
  float* st = stg + winb * 16 * STG;

  // Per-lane channel constants (channel c = 16t + lm), BN folded to scale/shift.
  float bpv[4], pscale[4], pshift[4], g1bias[4], gscale[4], gshift[4], g2bias[4];
  v2f wpB[4];
#pragma unroll
  for (int t = 0; t < 4; ++t) {
    int c = 16 * t + lm;
    bpv[t] = b_p[c];
    float ps = bnp_g[c] * (1.0f / sqrtf(bnp_v[c] + 1e-5f));
    pscale[t] = ps;
    pshift[t] = bnp_b[c] - bnp_m[c] * ps;
    g1bias[t] = bg1[c];
    float gs = bng_g[c] * (1.0f / sqrtf(bng_v[c] + 1e-5f));
    gscale[t] = gs;
    gshift[t] = bng_b[c] - bng_m[c] * gs;
    g2bias[t] = bg2[c];
    // B operand of the p-GEMM (Wp is 3x64, K padded to 4 with zeros)
    wpB[t].x = hi ? Wp[2 * CCH + c] : Wp[0 * CCH + c];
    wpB[t].y = hi ? 0.0f            : Wp[1 * CCH + c];
  }

  // Neighbor indices for the rows this half-wave owns (m = r + 8*hi).
  int idx8[8];
#pragma unroll
  for (int r = 0; r < 8; ++r)
    idx8[r] = gidx[(size_t)i * KNBR + r + 8 * hi];

  // rel as A operand (A-layout: lane m holds {x,y} low-half, {z,0} high-half).
  int   mi = gidx[(size_t)i * KNBR + lm];
  float nx = points[(size_t)mi * 3 + 0] - points[(size_t)i * 3 + 0];
  float ny = points[(size_t)mi * 3 + 1] - points[(size_t)i * 3 + 1];
  float nz = points[(size_t)mi * 3 + 2] - points[(size_t)i * 3 + 2];
  v2f relA;
  relA.x = hi ? nz : nx;
  relA.y = hi ? 0.0f : ny;

  // p = relu(bn(rel @ Wp + b_p))
  float p[4][8];
#pragma unroll
  for (int t = 0; t < 4; ++t) {
    v8f acc = {};
    acc = wmma_f32(relA, wpB[t], acc);
#pragma unroll
    for (int r = 0; r < 8; ++r) {
      float x = (acc[r] + bpv[t]) * pscale[t] + pshift[t];
      p[t][r] = x > 0.0f ? x : 0.0f;
    }
  }

  // a = q_i - k[idx] + p  -> staging (row-major, stride 68 floats)
#pragma unroll
  for (int t = 0; t < 4; ++t) {
    float qv = qb[(size_t)i * CCH + 16 * t + lm];
#pragma unroll
    for (int r = 0; r < 8; ++r) {
      float kv = kb[(size_t)idx8[r] * CCH + 16 * t + lm];
      st[(r + 8 * hi) * STG + 16 * t + lm] = qv - kv + p[t][r];
    }
  }
  asm volatile("s_wait_dscnt 0" ::: "memory");  // staging visible within wave

  // GEMM 1: t1 = relu(bn(a @ Wg1 + bg1))
  const float* arow = st + lm * STG + 2 * hi;   // A base, +16B per k-step
  float t1[4][8];
#pragma unroll
  for (int t = 0; t < 4; ++t) {
    const float* bbase = w1 + hi * 128 + (((16 * t + lm) ^ (hi << 4)) << 1);
    v8f acc = {};
#pragma unroll
    for (int s = 0; s < 16; ++s) {
      v2f av = *(const v2f*)(arow + 4 * s);        // ds_load_b64, imm offset
      v2f bv = *(const v2f*)(bbase + s * 256);     // ds_load_b64, imm offset
      acc = wmma_f32(av, bv, acc);
    }
#pragma unroll
    for (int r = 0; r < 8; ++r) {
      float x = (acc[r] + g1bias[t]) * gscale[t] + gshift[t];
      t1[t][r] = x > 0.0f ? x : 0.0f;
    }
  }

  // Overwrite staging with t1 (all tiles have finished reading `a`).
#pragma unroll
  for (int t = 0; t < 4; ++t)
#pragma unroll
    for (int r = 0; r < 8; ++r)
      st[(r + 8 * hi) * STG + 16 * t + lm] = t1[t][r];
  asm volatile("s_wait_dscnt 0" ::: "memory");

  // GEMM 2: h = t1 @ Wg2 + bg2
  float h[4][8];
#pragma unroll
  for (int t = 0; t < 4; ++t) {
    const float* bbase = w2 + hi * 128 + (((16 * t + lm) ^ (hi << 4)) << 1);
    v8f acc = {};
#pragma unroll
    for (int s = 0; s < 16; ++s) {
      v2f av = *(const v2f*)(arow + 4 * s);
      v2f bv = *(const v2f*)(bbase + s * 256);
      acc = wmma_f32(av, bv, acc);
    }
#pragma unroll
    for (int r = 0; r < 8; ++r)
      h[t][r] = acc[r] + g2bias[t];
  }

  // Softmax over the 16 neighbors per channel, then weighted sum with (v+p).
  // Channel c = 16t+lm lives in lanes lm and lm+16 (8 rows each).
#pragma unroll
  for (int t = 0; t < 4; ++t) {
    float mx = h[t][0];
#pragma unroll
    for (int r = 1; r < 8; ++r) mx = fmaxf(mx, h[t][r]);
    mx = fmaxf(mx, __shfl_xor(mx, 16, 32));
    float ssum = 0.0f;
#pragma unroll
    for (int r = 0; r < 8; ++r) {
      float e = __expf(h[t][r] - mx);
      h[t][r] = e;
      ssum += e;
    }
    ssum += __shfl_xor(ssum, 16, 32);
    float inv = 1.0f / ssum;

    float acc = 0.0f;
#pragma unroll
    for (int r = 0; r < 8; ++r) {
      float gvv = vb[(size_t)idx8[r] * CCH + 16 * t + lm];
      acc += (gvv + p[t][r]) * h[t][r];
    }
    acc *= inv;
    acc += __shfl_xor(acc, 16, 32);
    if (hi == 0) sb[(size_t)i * CCH + 16 * t + lm] = acc;
  }
}

// ---------------------------------------------------------------------------
// Kernel 3: out = s @ Wo + bo.  Same tiling as kernel 1, single matrix.
// ---------------------------------------------------------------------------
__global__ __launch_bounds__(256) void proj_kernel(
    const float* __restrict__ sbuf, const float* __restrict__ Wo,
    const float* __restrict__ bo, float* __restrict__ out, int ntiles)
{
  __shared__ float ldsW[CCH * CCH];
  load_w_swz(ldsW, Wo);
  __syncthreads();

  int wave = blockIdx.x * WAVES_PER_BLOCK + (threadIdx.x >> 5);
  if (wave >= ntiles) return;
  int lane = threadIdx.x & 31;
  int hi = lane >> 4, lm = lane & 15;
  int base = wave * 16;

  v2f A[16];
  const float* arow = sbuf + (size_t)(base + lm) * CCH;
#pragma unroll
  for (int s = 0; s < 16; ++s)
    A[s] = *(const v2f*)(arow + 4 * s + 2 * hi);

#pragma unroll
  for (int t = 0; t < 4; ++t) {
    const float* bbase = ldsW + hi * 128 + (((16 * t + lm) ^ (hi << 4)) << 1);
    v8f acc = {};
#pragma unroll
    for (int s = 0; s < 16; ++s) {
      v2f b = *(const v2f*)(bbase + s * 256);
      acc = wmma_f32(A[s], b, acc);
    }
    float bias = bo[16 * t + lm];
#pragma unroll
    for (int r = 0; r < 8; ++r)
      out[(size_t)(base + r + 8 * hi) * CCH + 16 * t + lm] = acc[r] + bias;
  }
}

// ---------------------------------------------------------------------------
extern "C" void kernel_launch(void* const* d_in, const int* in_sizes, int n_in,
                              void* d_out, int out_size, void* d_ws, size_t ws_size,
                              hipStream_t stream)
{
  const float* points   = (const float*)d_in[0];
  const float* features = (const float*)d_in[1];
  const int*   gidx     = (const int*)d_in[2];
  const float* Wq  = (const float*)d_in[3];  const float* bq  = (const float*)d_in[4];
  const float* Wk  = (const float*)d_in[5];  const float* bk  = (const float*)d_in[6];
  const float* Wv  = (const float*)d_in[7];  const float* bv  = (const float*)d_in[8];
  const float* Wp  = (const float*)d_in[9];  const float* b_p = (const float*)d_in[10];
  const float* bnp_g = (const float*)d_in[11]; const float* bnp_b = (const float*)d_in[12];
  const float* bnp_m = (const float*)d_in[13]; const float* bnp_v = (const float*)d_in[14];
  const float* Wg1 = (const float*)d_in[15]; const float* bg1 = (const float*)d_in[16];
  const float* bng_g = (const float*)d_in[17]; const float* bng_b = (const float*)d_in[18];
  const float* bng_m = (const float*)d_in[19]; const float* bng_v = (const float*)d_in[20];
  const float* Wg2 = (const float*)d_in[21]; const float* bg2 = (const float*)d_in[22];
  const float* Wo  = (const float*)d_in[23]; const float* bo  = (const float*)d_in[24];

  const int N = in_sizes[0] / 3;        // points is (N, 3); N = 50000 (16-aligned)
  float* qb = (float*)d_ws;             // workspace: q | k | v | s, each N*64 f32
  float* kb = qb + (size_t)N * CCH;
  float* vb = kb + (size_t)N * CCH;
  float* sb = vb + (size_t)N * CCH;

  const int ntiles = (N + 15) / 16;
  const int nblk_t = (ntiles + WAVES_PER_BLOCK - 1) / WAVES_PER_BLOCK;
  const int nblk_p = (N + WAVES_PER_BLOCK - 1) / WAVES_PER_BLOCK;

  qkv_kernel<<<nblk_t, 256, 0, stream>>>(features, Wq, bq, Wk, bk, Wv, bv,
                                         qb, kb, vb, ntiles);
  pt_main_kernel<<<nblk_p, 256, 0, stream>>>(points, gidx, qb, kb, vb,
                                             Wp, b_p, bnp_g, bnp_b, bnp_m, bnp_v,
                                             Wg1, bg1, bng_g, bng_b, bng_m, bng_v,
                                             Wg2, bg2, sb, N);
  proj_kernel<<<nblk_t, 256, 0, stream>>>(sb, Wo, bo, (float*)d_out, ntiles);
}